// SparsityLearning_32985348833730
// MI455X (gfx1250) — compile-verified
//
#include <hip/hip_runtime.h>
#include <math.h>

#define D 128
#define TEMP 0.5f
#define THRESH 0.5f
#define FEPS 1e-6f

typedef __attribute__((ext_vector_type(2))) float v2f;
typedef __attribute__((ext_vector_type(8))) float v8f;

#if defined(__AMDGCN__) && __has_builtin(__builtin_amdgcn_global_load_async_to_lds_b128) && \
    __has_builtin(__builtin_amdgcn_s_wait_asynccnt)
#define USE_ASYNC_LDS 1
typedef int v4i __attribute__((ext_vector_type(4)));
typedef __attribute__((address_space(1))) v4i* gv4p;   // global (AS1) int4*
typedef __attribute__((address_space(3))) v4i* lv4p;   // LDS (AS3) int4*
#else
#define USE_ASYNC_LDS 0
#endif

__device__ __forceinline__ float wave_max(float v) {
#pragma unroll
  for (int off = 16; off > 0; off >>= 1) v = fmaxf(v, __shfl_xor(v, off, 32));
  return v;
}
__device__ __forceinline__ float wave_sum(float v) {
#pragma unroll
  for (int off = 16; off > 0; off >>= 1) v += __shfl_xor(v, off, 32);
  return v;
}

// ---------------- 0: zero counters ----------------
__global__ void k_zero(float* deg, int* cnt, int* cursor, int n) {
  int i = blockIdx.x * blockDim.x + threadIdx.x;
  if (i < n) { deg[i] = 0.f; cnt[i] = 0; cursor[i] = 0; }
}

// ---------------- 1: xw = x @ W via v_wmma_f32_16x16x4_f32 ----------------
// One wave computes a 16x16 tile of xw. W (64KB fp32) staged in LDS via the
// gfx1250 async global->LDS path (ASYNCcnt) when available.
__global__ void k_gemm_xw(const float* __restrict__ x, const float* __restrict__ W,
                          float* __restrict__ xw, int nrows) {
  __shared__ float sW[D * D];
#if USE_ASYNC_LDS
  {
    gv4p gW = (gv4p)W;
    lv4p lW = (lv4p)sW;
    for (int i = threadIdx.x; i < (D * D) / 4; i += blockDim.x)
      __builtin_amdgcn_global_load_async_to_lds_b128(gW + i, lW + i, 0, 0);
    __builtin_amdgcn_s_wait_asynccnt(0);
  }
#else
  for (int i = threadIdx.x; i < D * D; i += blockDim.x) sW[i] = W[i];
#endif
  __syncthreads();

  const int wave = threadIdx.x >> 5;
  const int lane = threadIdx.x & 31;
  const int tile = blockIdx.x * (blockDim.x >> 5) + wave;
  const int tiles_n = D / 16;                 // 8
  const int tm = tile / tiles_n, tn = tile % tiles_n;
  const int row0 = tm * 16;
  if (row0 >= nrows) return;

  const int half = lane >> 4;                 // 0: K pair {0,1}, 1: K pair {2,3}
  const int l15  = lane & 15;
  const bool full = (row0 + 16 <= nrows);
  const int arow = full ? (row0 + l15) : min(row0 + l15, nrows - 1);
  const int bcol = tn * 16 + l15;

  v8f c = {};
#pragma unroll 4
  for (int k = 0; k < D; k += 4) {
    const int ka = k + half * 2;
    v2f a, b;
    // A 16x4 (MxK): lanes 0-15 -> K={k,k+1}, lanes 16-31 -> K={k+2,k+3}
    a.x = x[(size_t)arow * D + ka];
    a.y = x[(size_t)arow * D + ka + 1];
    // B 4x16 (KxN): rows striped across lanes within a VGPR (mirror of C layout)
    b.x = sW[ka * D + bcol];
    b.y = sW[(ka + 1) * D + bcol];
    c = __builtin_amdgcn_wmma_f32_16x16x4_f32(false, a, false, b, (short)0, c,
                                              false, false);
  }
  // C/D 16x16: lanes 0-15 -> M=v, lanes 16-31 -> M=v+8; N = lane&15
  float* dst = xw + (size_t)(row0 + half * 8) * D + tn * 16 + l15;
  if (full) {
#pragma unroll
    for (int v = 0; v < 8; ++v) dst[(size_t)v * D] = c[v];
  } else {
#pragma unroll
    for (int v = 0; v < 8; ++v)
      if (row0 + v + half * 8 < nrows) dst[(size_t)v * D] = c[v];
  }
}

// ---------------- 2: degree + destination histogram ----------------
__global__ void k_deg_hist(const int* __restrict__ row, const int* __restrict__ col,
                           const float* __restrict__ ew, float* deg, int* cnt, int E) {
  int e = blockIdx.x * blockDim.x + threadIdx.x;
  if (e < E) {
    atomicAdd(&deg[row[e]], ew[e]);
    atomicAdd(&cnt[col[e]], 1);
  }
}

// ---------------- 3: dinv = (deg+1)^-0.5 ----------------
__global__ void k_dinv(const float* deg, float* dinv, int n) {
  int i = blockIdx.x * blockDim.x + threadIdx.x;
  if (i < n) dinv[i] = rsqrtf(deg[i] + 1.0f);
}

// ---------------- 4: agg = dinv^2 * xw  (self-loop message) ----------------
__global__ void k_init_agg(const float* __restrict__ xw, const float* __restrict__ dinv,
                           float* __restrict__ agg, int n4) {   // n4 = N*D/4
  int i = blockIdx.x * blockDim.x + threadIdx.x;
  if (i >= n4) return;
  int node = i >> 5;                  // D/4 == 32 float4 per row
  float s = dinv[node]; s *= s;
  float4 v = ((const float4*)xw)[i];
  v.x *= s; v.y *= s; v.z *= s; v.w *= s;
  ((float4*)agg)[i] = v;
}

// ---------------- 5: edge aggregation (one wave per edge) ----------------
__global__ void k_edge_agg(const int* __restrict__ row, const int* __restrict__ col,
                           const float* __restrict__ ew, const float* __restrict__ dinv,
                           const float* __restrict__ xw, float* __restrict__ agg, int E) {
  int wave = (blockIdx.x * blockDim.x + threadIdx.x) >> 5;
  int lane = threadIdx.x & 31;
  if (wave >= E) return;
  int r = row[wave], c = col[wave];
  float coef = dinv[r] * ew[wave] * dinv[c];
  if (coef == 0.f) return;            // masked edges (half of them) move no bytes
  float4 v = ((const float4*)(xw + (size_t)r * D))[lane];
  float* dst = agg + (size_t)c * D + lane * 4;
  atomicAdd(dst + 0, coef * v.x);
  atomicAdd(dst + 1, coef * v.y);
  atomicAdd(dst + 2, coef * v.z);
  atomicAdd(dst + 3, coef * v.w);
}

// ---------------- 6: xg = relu(agg) in place; a_src/a_dst dots ----------------
__global__ void k_relu_att(float* __restrict__ agg, const float* __restrict__ att,
                           float* __restrict__ asrc, float* __restrict__ adst, int n) {
  int wave = (blockIdx.x * blockDim.x + threadIdx.x) >> 5;
  int lane = threadIdx.x & 31;
  if (wave >= n) return;
  float4 a0 = ((const float4*)att)[lane];        // att[0:128]
  float4 a1 = ((const float4*)att)[32 + lane];   // att[128:256]
  float4* xp = (float4*)(agg + (size_t)wave * D);
  float4 v = xp[lane];
  v.x = fmaxf(v.x, 0.f); v.y = fmaxf(v.y, 0.f);
  v.z = fmaxf(v.z, 0.f); v.w = fmaxf(v.w, 0.f);
  xp[lane] = v;
  float s = v.x * a0.x + v.y * a0.y + v.z * a0.z + v.w * a0.w;
  float d = v.x * a1.x + v.y * a1.y + v.z * a1.z + v.w * a1.w;
  s = wave_sum(s); d = wave_sum(d);
  if (lane == 0) { asrc[wave] = s; adst[wave] = d; }
}

// ---------------- 7: hierarchical exclusive scan of (cnt[i]+1) ----------------
__global__ void k_scan_block(const int* __restrict__ cnt, int* __restrict__ startArr,
                             int* __restrict__ blockSums, int n) {
  __shared__ int tmp[256];
  int t = threadIdx.x, b = blockIdx.x;
  int base = b * 1024 + t * 4;
  int v[4]; int s = 0;
#pragma unroll
  for (int j = 0; j < 4; ++j) {
    int idx = base + j;
    v[j] = (idx < n) ? (cnt[idx] + 1) : 0;
    s += v[j];
  }
  tmp[t] = s;
  __syncthreads();
  for (int off = 1; off < 256; off <<= 1) {
    int add = (t >= off) ? tmp[t - off] : 0;
    __syncthreads();
    tmp[t] += add;
    __syncthreads();
  }
  int run = tmp[t] - s;                // block-local exclusive base
#pragma unroll
  for (int j = 0; j < 4; ++j) {
    int idx = base + j;
    if (idx < n) { startArr[idx] = run; run += v[j]; }
  }
  if (t == 255) blockSums[b] = tmp[255];
}
__global__ void k_scan_sums(int* blockSums, int nb) {
  if (threadIdx.x == 0 && blockIdx.x == 0) {
    int run = 0;
    for (int b = 0; b < nb; ++b) { int s = blockSums[b]; blockSums[b] = run; run += s; }
  }
}
__global__ void k_scan_add(int* startArr, const int* blockSums, int n) {
  int i = blockIdx.x * blockDim.x + threadIdx.x;
  if (i < n) startArr[i] += blockSums[i >> 10];
}

// ---------------- 8: counting-sort scatter (original edges) ----------------
__global__ void k_scatter_edges(const int* __restrict__ row, const int* __restrict__ col,
                                const float* __restrict__ ew, const float* __restrict__ em,
                                const float* __restrict__ asrc, const float* __restrict__ adst,
                                const int* __restrict__ startArr, int* __restrict__ cursor,
                                float* out_row, float* out_col, float* out_ew,
                                float* out_em, float* wsrt, int E) {
  int e = blockIdx.x * blockDim.x + threadIdx.x;
  if (e >= E) return;
  int r = row[e], c = col[e];
  float w = asrc[r] + adst[c];
  w = (w > 0.f) ? w : 0.01f * w;       // leaky_relu(0.01)
  w += ew[e];
  int pos = startArr[c] + atomicAdd(&cursor[c], 1);
  out_row[pos] = (float)r;
  out_col[pos] = (float)c;
  out_ew[pos]  = ew[e];
  out_em[pos]  = em[e];
  wsrt[pos]    = w;
}

// ---------------- 9: scatter self loops (last in each bucket) ----------------
__global__ void k_scatter_self(const float* __restrict__ asrc, const float* __restrict__ adst,
                               const int* __restrict__ startArr, const int* __restrict__ cnt,
                               float* out_row, float* out_col, float* out_ew,
                               float* out_em, float* wsrt, int n) {
  int c = blockIdx.x * blockDim.x + threadIdx.x;
  if (c >= n) return;
  float w = asrc[c] + adst[c];
  w = (w > 0.f) ? w : 0.01f * w;       // +0 weight for self loops
  int pos = startArr[c] + cnt[c];
  out_row[pos] = (float)c;
  out_col[pos] = (float)c;
  out_ew[pos]  = 0.f;
  out_em[pos]  = -1.f;
  wsrt[pos]    = w;
}

// ---------------- 10: segment softmax + relaxed Bernoulli (one wave/node) ----
__global__ void k_softmax_sample(const float* __restrict__ wsrt,
                                 const int* __restrict__ startArr, const int* __restrict__ cnt,
                                 const float* __restrict__ noise, const int* __restrict__ layerp,
                                 float* out_ew, float* out_ys, float* out_em,
                                 float* out_intra, int n) {
  int node = (blockIdx.x * blockDim.x + threadIdx.x) >> 5;
  int lane = threadIdx.x & 31;
  if (node >= n) return;
  int s   = startArr[node];
  int len = cnt[node] + 1;
  float layerp1 = (float)(layerp[0]) + 1.0f;

  float mx = -INFINITY;
  for (int i = lane; i < len; i += 32) mx = fmaxf(mx, wsrt[s + i]);
  mx = wave_max(mx);
  float den = 0.f;
  for (int i = lane; i < len; i += 32) den += __expf(wsrt[s + i] - mx);
  den = wave_sum(den);

  for (int i = lane; i < len; i += 32) {
    int pos = s + i;
    float p = __expf(wsrt[pos] - mx) / den;
    p = fminf(fmaxf(p, FEPS), 1.f - FEPS);
    float u = fminf(fmaxf(noise[pos], FEPS), 1.f - FEPS);
    float logits = logf(p) - log1pf(-p) + logf(u) - log1pf(-u);
    float ys = 1.f / (1.f + __expf(-logits / TEMP));
    float y  = (ys > THRESH) ? 1.f : 0.f;       // straight-through forward value
    out_ys[pos] = ys;
    float ews = out_ew[pos];
    out_ew[pos] = (ews == 0.f) ? y : ews;
    float ems = out_em[pos];
    out_em[pos] = (ems == 0.f && y == 1.f) ? layerp1 : ems;
    if (ems == -1.f) out_intra[node] = ys;       // self loop: sorted by node id
  }
}

extern "C" void kernel_launch(void* const* d_in, const int* in_sizes, int n_in,
                              void* d_out, int out_size, void* d_ws, size_t ws_size,
                              hipStream_t stream) {
  const float* x    = (const float*)d_in[0];
  const int*   ei   = (const int*)d_in[1];
  const float* ew   = (const float*)d_in[2];
  const float* em   = (const float*)d_in[3];
  const int*   layr = (const int*)d_in[4];
  const float* att  = (const float*)d_in[5];
  const float* W    = (const float*)d_in[6];
  const float* nois = (const float*)d_in[7];

  const int N  = in_sizes[0] / D;      // 100000
  const int E  = in_sizes[2];          // 1600000
  const int EN = E + N;
  const int* row = ei;
  const int* col = ei + E;

  // ---- workspace layout (~113 MB) ----
  float* xw   = (float*)d_ws;
  float* agg  = xw   + (size_t)N * D;
  float* dinv = agg  + (size_t)N * D;
  float* asrc = dinv + N;
  float* adst = asrc + N;
  float* deg  = adst + N;
  int* cnt     = (int*)(deg + N);
  int* startA  = cnt + N;
  int* cursor  = startA + N;
  int* blkSums = cursor + N;           // up to a few hundred entries
  float* wsrt  = (float*)(blkSums + 512);

  // ---- output layout (float*, ints stored as exact float values) ----
  float* out      = (float*)d_out;
  float* out_row  = out;
  float* out_col  = out + (size_t)EN;
  float* out_ew   = out + (size_t)2 * EN;
  float* out_ys   = out + (size_t)3 * EN;
  float* out_em   = out + (size_t)4 * EN;
  float* out_intr = out + (size_t)5 * EN;

  const int B = 256;
  // 0: zero counters
  k_zero<<<(N + B - 1) / B, B, 0, stream>>>(deg, cnt, cursor, N);
  // 1: WMMA GEMM
  {
    int tiles  = ((N + 15) / 16) * (D / 16);
    int blocks = (tiles + 7) / 8;      // 8 waves per block
    k_gemm_xw<<<blocks, B, 0, stream>>>(x, W, xw, N);
  }
  // 2: degree + histogram
  k_deg_hist<<<(E + B - 1) / B, B, 0, stream>>>(row, col, ew, deg, cnt, E);
  // 3: dinv
  k_dinv<<<(N + B - 1) / B, B, 0, stream>>>(deg, dinv, N);
  // 4: self-loop message
  {
    int n4 = N * (D / 4);
    k_init_agg<<<(n4 + B - 1) / B, B, 0, stream>>>(xw, dinv, agg, n4);
  }
  // 5: edge aggregation (wave per edge)
  {
    long long th = (long long)E * 32;
    k_edge_agg<<<(int)((th + B - 1) / B), B, 0, stream>>>(row, col, ew, dinv, xw, agg, E);
  }
  // 6: relu + attention dots (wave per node)
  {
    long long th = (long long)N * 32;
    k_relu_att<<<(int)((th + B - 1) / B), B, 0, stream>>>(agg, att, asrc, adst, N);
  }
  // 7: exclusive scan of bucket sizes (cnt+1)
  {
    int nb = (N + 1023) / 1024;
    k_scan_block<<<nb, 256, 0, stream>>>(cnt, startA, blkSums, N);
    k_scan_sums<<<1, 32, 0, stream>>>(blkSums, nb);
    k_scan_add<<<(N + B - 1) / B, B, 0, stream>>>(startA, blkSums, N);
  }
  // 8/9: counting-sort scatter
  k_scatter_edges<<<(E + B - 1) / B, B, 0, stream>>>(row, col, ew, em, asrc, adst,
                                                     startA, cursor, out_row, out_col,
                                                     out_ew, out_em, wsrt, E);
  k_scatter_self<<<(N + B - 1) / B, B, 0, stream>>>(asrc, adst, startA, cnt, out_row,
                                                    out_col, out_ew, out_em, wsrt, N);
  // 10: segment softmax + relaxed Bernoulli + output fixups
  {
    long long th = (long long)N * 32;
    k_softmax_sample<<<(int)((th + B - 1) / B), B, 0, stream>>>(
        wsrt, startA, cnt, nois, layr, out_ew, out_ys, out_em, out_intr, N);
  }
}